// GraphTripleConv_29996051595776
// MI455X (gfx1250) — compile-verified
//
#include <hip/hip_runtime.h>

#define D_IN  128
#define H_DIM 512
#define D_OUTC 128

typedef __attribute__((ext_vector_type(2))) float v2f;
typedef __attribute__((ext_vector_type(8))) float v8f;

// One wave computes a 16x16 fp32 tile: C = A(16xK, LDS) * B(KxN, global), K step 4.
// A lane layout (ISA 7.12.2, 32-bit A 16x4): lanes 0-15 -> K={kk,kk+1}, lanes 16-31 -> K={kk+2,kk+3}, M=lane&15.
// B symmetric: N=lane&15, same K pairing. C/D: col = lane&15, rows = v + 8*(lane>>4).
__device__ __forceinline__ v8f wmma_tile_f32(const float* __restrict__ Alds, int lda,
                                             const float* __restrict__ B, int ldb,
                                             int K, int n, int lane) {
    const int m  = lane & 15;
    const int kh = (lane >> 4) << 1;        // 0 or 2
    v8f c = {};
    for (int kk = 0; kk < K; kk += 4) {
        const int k0 = kk + kh;
        v2f a, b;
        a.x = Alds[m * lda + k0];
        a.y = Alds[m * lda + k0 + 1];
        b.x = B[(size_t)k0 * ldb + n];
        b.y = B[(size_t)(k0 + 1) * ldb + n];
        c = __builtin_amdgcn_wmma_f32_16x16x4_f32(false, a, false, b, (short)0, c, false, false);
    }
    return c;
}

// ---------------- GEMM1: h = relu([obj[s] | pred | obj[o]] @ W1 + b1) ----------------
__global__ __launch_bounds__(256) void k_gemm1(const float* __restrict__ obj,
                                               const float* __restrict__ pred,
                                               const int* __restrict__ edges,
                                               const float* __restrict__ W1,
                                               const float* __restrict__ b1,
                                               float* __restrict__ h) {
    const int K   = 3 * D_IN;       // 384
    const int LDA = K + 4;          // pad: conflict-free, rows stay 16B aligned
    __shared__ __align__(16) float Alds[16 * (3 * D_IN + 4)];

    const int tid = threadIdx.x;
    const int t0  = blockIdx.x * 16;

    // Gather-concat LDS fill: 16 rows x 96 float4 = 1536 chunks, 6 per thread, no divergence.
    for (int idx = tid; idx < 16 * (K / 4); idx += 256) {
        const int r   = idx / (K / 4);
        const int c4  = idx % (K / 4);
        const int t   = t0 + r;
        const int seg = c4 >> 5;            // 0: obj[s], 1: pred[t], 2: obj[o]
        const int off = c4 & 31;
        const float4* src;
        if (seg == 0)      src = (const float4*)(obj  + (size_t)edges[2 * t]     * D_IN);
        else if (seg == 1) src = (const float4*)(pred + (size_t)t                * D_IN);
        else               src = (const float4*)(obj  + (size_t)edges[2 * t + 1] * D_IN);
        ((float4*)(Alds + r * LDA))[c4] = src[off];
    }
    __syncthreads();

    const int lane = tid & 31;
    const int wave = tid >> 5;
    const int n    = blockIdx.y * 128 + wave * 16 + (lane & 15);

    v8f c = wmma_tile_f32(Alds, LDA, W1, H_DIM, K, n, lane);

    const int   mb   = (lane >> 4) << 3;
    const float bias = b1[n];
    for (int v = 0; v < 8; ++v) {
        float val = c[v] + bias;
        h[(size_t)(t0 + mb + v) * H_DIM + n] = val > 0.f ? val : 0.f;
    }
}

// ---------------- GEMM2: new_t = relu(h @ W2 + b2), routed epilogue ----------------
__global__ __launch_bounds__(256) void k_gemm2(const float* __restrict__ h,
                                               const float* __restrict__ W2,
                                               const float* __restrict__ b2,
                                               const int* __restrict__ edges,
                                               float* __restrict__ pooled,
                                               float* __restrict__ new_p) {
    const int K   = H_DIM;          // 512
    const int LDA = K + 4;
    const int N   = 2 * H_DIM + D_OUTC;  // 1152
    __shared__ __align__(16) float Alds[16 * (H_DIM + 4)];

    const int tid = threadIdx.x;
    const int t0  = blockIdx.x * 16;

    for (int idx = tid; idx < 16 * (K / 4); idx += 256) {
        const int r  = idx >> 7;    // K/4 == 128
        const int c4 = idx & 127;
        ((float4*)(Alds + r * LDA))[c4] =
            ((const float4*)(h + (size_t)(t0 + r) * K))[c4];
    }
    __syncthreads();

    const int lane = tid & 31;
    const int wave = tid >> 5;
    const int n    = blockIdx.y * 128 + wave * 16 + (lane & 15);   // [0, 1152)

    v8f c = wmma_tile_f32(Alds, LDA, W2, N, K, n, lane);

    const int   mb   = (lane >> 4) << 3;
    const float bias = b2[n];
    for (int v = 0; v < 8; ++v) {
        float val = c[v] + bias;
        val = val > 0.f ? val : 0.f;
        const int t = t0 + mb + v;
        if (n < H_DIM) {                               // new_s -> pooled[s_idx]
            atomicAdd(pooled + (size_t)edges[2 * t] * H_DIM + n, val);
        } else if (n < H_DIM + D_OUTC) {               // new_p
            new_p[(size_t)t * D_OUTC + (n - H_DIM)] = val;
        } else {                                       // new_o -> pooled[o_idx]
            atomicAdd(pooled + (size_t)edges[2 * t + 1] * H_DIM + (n - H_DIM - D_OUTC), val);
        }
    }
}

// ---------------- counts[s] += 1, counts[o] += 1 ----------------
__global__ void k_counts(const int* __restrict__ edges, float* __restrict__ counts, int T) {
    const int t = blockIdx.x * blockDim.x + threadIdx.x;
    if (t < T) {
        atomicAdd(counts + edges[2 * t],     1.0f);
        atomicAdd(counts + edges[2 * t + 1], 1.0f);
    }
}

// ---------------- GEMM over objects: C = relu((A [* 1/max(cnt,1)]) @ B + bias) ----------------
__global__ __launch_bounds__(256) void k_gemm_obj(const float* __restrict__ A,
                                                  const float* __restrict__ counts,  // nullable
                                                  const float* __restrict__ B,
                                                  const float* __restrict__ bias_v,
                                                  float* __restrict__ C, int N) {
    const int K   = H_DIM;
    const int LDA = K + 4;
    __shared__ __align__(16) float Alds[16 * (H_DIM + 4)];

    const int tid = threadIdx.x;
    const int m0  = blockIdx.x * 16;

    for (int idx = tid; idx < 16 * (K / 4); idx += 256) {
        const int r  = idx >> 7;
        const int c4 = idx & 127;
        float4 val = ((const float4*)(A + (size_t)(m0 + r) * K))[c4];
        if (counts) {   // uniform branch: fuse mean-pool normalization into the LDS fill
            const float s = 1.0f / fmaxf(counts[m0 + r], 1.0f);
            val.x *= s; val.y *= s; val.z *= s; val.w *= s;
        }
        ((float4*)(Alds + r * LDA))[c4] = val;
    }
    __syncthreads();

    const int lane = tid & 31;
    const int wave = tid >> 5;
    const int n    = blockIdx.y * 128 + wave * 16 + (lane & 15);

    v8f c = wmma_tile_f32(Alds, LDA, B, N, K, n, lane);

    const int   mb = (lane >> 4) << 3;
    const float bb = bias_v[n];
    for (int v = 0; v < 8; ++v) {
        float val = c[v] + bb;
        C[(size_t)(m0 + mb + v) * N + n] = val > 0.f ? val : 0.f;
    }
}

extern "C" void kernel_launch(void* const* d_in, const int* in_sizes, int n_in,
                              void* d_out, int out_size, void* d_ws, size_t ws_size,
                              hipStream_t stream) {
    const float* obj   = (const float*)d_in[0];
    const float* pred  = (const float*)d_in[1];
    const int*   edges = (const int*)  d_in[2];
    const float* W1 = (const float*)d_in[3];
    const float* b1 = (const float*)d_in[4];
    const float* W2 = (const float*)d_in[5];
    const float* b2 = (const float*)d_in[6];
    const float* W3 = (const float*)d_in[7];
    const float* b3 = (const float*)d_in[8];
    const float* W4 = (const float*)d_in[9];
    const float* b4 = (const float*)d_in[10];

    const int O = in_sizes[0] / D_IN;   // 100000
    const int T = in_sizes[1] / D_IN;   // 200000

    float* new_obj = (float*)d_out;                                 // (O, 128)
    float* new_p   = (float*)d_out + (size_t)O * D_OUTC;            // (T, 128)

    uint8_t* ws = (uint8_t*)d_ws;
    size_t off = 0;
    float* h_buf  = (float*)(ws + off); off += (size_t)T * H_DIM * sizeof(float);
    float* pooled = (float*)(ws + off); off += (size_t)O * H_DIM * sizeof(float);
    float* counts = (float*)(ws + off); off += (size_t)O * sizeof(float);
    float* tmp    = (float*)(ws + off); off += (size_t)O * H_DIM * sizeof(float);

    // pooled + counts are contiguous: one capture-legal async memset.
    hipMemsetAsync(pooled, 0, ((size_t)O * H_DIM + (size_t)O) * sizeof(float), stream);

    k_counts<<<(T + 255) / 256, 256, 0, stream>>>(edges, counts, T);

    k_gemm1<<<dim3(T / 16, H_DIM / 128), 256, 0, stream>>>(obj, pred, edges, W1, b1, h_buf);

    k_gemm2<<<dim3(T / 16, (2 * H_DIM + D_OUTC) / 128), 256, 0, stream>>>(
        h_buf, W2, b2, edges, pooled, new_p);

    k_gemm_obj<<<dim3(O / 16, H_DIM / 128), 256, 0, stream>>>(pooled, counts, W3, b3, tmp, H_DIM);

    k_gemm_obj<<<dim3(O / 16, D_OUTC / 128), 256, 0, stream>>>(tmp, nullptr, W4, b4, new_obj, D_OUTC);
}